// GATGraphEmbedding_11553462026424
// MI455X (gfx1250) — compile-verified
//
#include <hip/hip_runtime.h>
#include <hip/hip_bf16.h>

// ---------------- problem constants (match reference) ----------------
#define N_NODES  50000
#define E_ORIG   800000
#define E_TOT    (E_ORIG + N_NODES)   // self-loops appended
#define IN_DIM   128
#define HID      64
#define H1       4
#define C1       (H1 * HID)           // 256
#define N_GRAPHS 64
#define SLOPE    0.2f
#define SCAN_B   256

typedef float v2f __attribute__((ext_vector_type(2)));
typedef float v8f __attribute__((ext_vector_type(8)));

// src/dst with implicit self-loops
__device__ __forceinline__ int edge_src(const int* ei, int e) {
  return (e < E_ORIG) ? ei[e] : (e - E_ORIG);
}
__device__ __forceinline__ int edge_dst(const int* ei, int e) {
  return (e < E_ORIG) ? ei[E_ORIG + e] : (e - E_ORIG);
}

// ---------------- utility fills / copy ----------------
__global__ void fill_f32(float* __restrict__ p, int n, float v) {
  int t = blockIdx.x * blockDim.x + threadIdx.x;
  if (t < n) p[t] = v;
}
__global__ void fill_i32(int* __restrict__ p, int n, int v) {
  int t = blockIdx.x * blockDim.x + threadIdx.x;
  if (t < n) p[t] = v;
}
__global__ void copy_i32(int* __restrict__ d, const int* __restrict__ s, int n) {
  int t = blockIdx.x * blockDim.x + threadIdx.x;
  if (t < n) d[t] = s[t];
}

// ---------------- fp32 WMMA GEMM: C[M,N] = A[M,K] * B[K,N] ----------------
// grid.x = M/16 ; blockDim.x = 32*(N/16) ; dyn-LDS = 16*K*4 B
// Each wave computes one 16x16 tile with V_WMMA_F32_16X16X4_F32.
__global__ void gemm_wmma_f32(const float* __restrict__ A, const float* __restrict__ B,
                              float* __restrict__ C, int K, int N) {
  extern __shared__ float Alds[];                 // [16][K]
  const int rowbase = blockIdx.x * 16;
  for (int i = threadIdx.x; i < 16 * K; i += blockDim.x) {
    int r = i / K, k = i - r * K;
    Alds[i] = A[(size_t)(rowbase + r) * K + k];
  }
  __syncthreads();

  const int lane = threadIdx.x & 31;
  const int wave = threadIdx.x >> 5;
  const int m    = lane & 15;
  const int kk   = (lane >> 4) << 1;              // 0 (lanes 0-15) or 2 (lanes 16-31)
  const int col  = wave * 16 + m;

  v8f acc = {};
  for (int k0 = 0; k0 < K; k0 += 4) {
    v2f a, b;
    a.x = Alds[m * K + k0 + kk];
    a.y = Alds[m * K + k0 + kk + 1];
    b.x = B[(size_t)(k0 + kk)     * N + col];
    b.y = B[(size_t)(k0 + kk + 1) * N + col];
    acc = __builtin_amdgcn_wmma_f32_16x16x4_f32(false, a, false, b,
                                                (short)0, acc, false, false);
  }
  const int rofs = (lane >> 4) << 3;              // 0 or 8
#pragma unroll
  for (int j = 0; j < 8; ++j)
    C[(size_t)(rowbase + j + rofs) * N + col] = acc[j];
}

// ---------------- CSR-by-dst build ----------------
__global__ void csr_deg(const int* __restrict__ ei, int* __restrict__ deg) {
  int e = blockIdx.x * blockDim.x + threadIdx.x;
  if (e >= E_TOT) return;
  atomicAdd(&deg[edge_dst(ei, e)], 1);
}

// block-local exclusive scan; emits per-block totals
__global__ void scan1(const int* __restrict__ in, int* __restrict__ out,
                      int* __restrict__ bsum, int n) {
  __shared__ int sh[SCAN_B];
  int t = threadIdx.x, g = blockIdx.x * SCAN_B + t;
  int v = (g < n) ? in[g] : 0;
  sh[t] = v;
  __syncthreads();
  for (int off = 1; off < SCAN_B; off <<= 1) {
    int add = (t >= off) ? sh[t - off] : 0;
    __syncthreads();
    sh[t] += add;
    __syncthreads();
  }
  if (g < n) out[g] = sh[t] - v;                  // exclusive within block
  if (t == SCAN_B - 1) bsum[blockIdx.x] = sh[t];  // block total
}
// single-block exclusive scan of block totals (nb <= 1024)
__global__ void scan2(int* __restrict__ bsum, int nb) {
  __shared__ int sh[1024];
  int t = threadIdx.x;
  int v = (t < nb) ? bsum[t] : 0;
  sh[t] = v;
  __syncthreads();
  for (int off = 1; off < 1024; off <<= 1) {
    int add = (t >= off) ? sh[t - off] : 0;
    __syncthreads();
    sh[t] += add;
    __syncthreads();
  }
  if (t < nb) bsum[t] = sh[t] - v;
}
__global__ void scan3(int* __restrict__ out, const int* __restrict__ bsum, int n, int total) {
  int g = blockIdx.x * SCAN_B + threadIdx.x;
  if (g < n) out[g] += bsum[blockIdx.x];
  if (g == 0) out[n] = total;                     // rowptr[N] = E_TOT
}

__global__ void csr_place(const int* __restrict__ ei, int* __restrict__ cursor,
                          int* __restrict__ csr_src) {
  int e = blockIdx.x * blockDim.x + threadIdx.x;
  if (e >= E_TOT) return;
  int dst = edge_dst(ei, e);
  int pos = atomicAdd(&cursor[dst], 1);
  csr_src[pos] = edge_src(ei, e);
}

// ---------------- per-(node,head) attention dots ----------------
__global__ void node_attn(const float* __restrict__ h, const float* __restrict__ asrc,
                          const float* __restrict__ adst, float* __restrict__ as_o,
                          float* __restrict__ ad_o, int H) {
  int tid = blockIdx.x * blockDim.x + threadIdx.x;
  if (tid >= N_NODES * H) return;
  int node = tid / H, head = tid - node * H;
  const float* hp = h + (size_t)node * H * HID + head * HID;
  const float* sp = asrc + head * HID;
  const float* dp = adst + head * HID;
  float ss = 0.f, dd = 0.f;
#pragma unroll 8
  for (int i = 0; i < HID; ++i) { float v = hp[i]; ss += v * sp[i]; dd += v * dp[i]; }
  as_o[tid] = ss;
  ad_o[tid] = dd;
}

// ---------------- segment softmax over CSR (no atomics) ----------------
// one thread per (dst, head): max -> exp/sum -> normalize its incoming edges
__global__ void attn_softmax_csr(const float* __restrict__ as_i, const float* __restrict__ ad_i,
                                 const int* __restrict__ rowptr, const int* __restrict__ csr_src,
                                 float* __restrict__ alpha, int H) {
  int tid = blockIdx.x * blockDim.x + threadIdx.x;
  if (tid >= N_NODES * H) return;
  int dst = tid / H, head = tid - dst * H;
  int s0 = rowptr[dst], s1 = rowptr[dst + 1];
  float adv = ad_i[dst * H + head];
  float m = -__builtin_inff();
  for (int p = s0; p < s1; ++p) {
    float e = as_i[csr_src[p] * H + head] + adv;
    e = (e > 0.f) ? e : SLOPE * e;
    m = fmaxf(m, e);
  }
  float sum = 0.f;
  for (int p = s0; p < s1; ++p) {
    float e = as_i[csr_src[p] * H + head] + adv;
    e = (e > 0.f) ? e : SLOPE * e;
    float v = __expf(e - m);
    alpha[p * H + head] = v;
    sum += v;
  }
  float inv = 1.f / sum;                           // self-loop guarantees sum > 0
  for (int p = s0; p < s1; ++p) alpha[p * H + head] *= inv;
}

// ---------------- gather aggregation + bias + ELU (no atomics) ----------------
// one thread per (dst, 4-channel chunk); 64 channels per head
__global__ void agg_csr_elu(const float* __restrict__ h, const float* __restrict__ alpha,
                            const int* __restrict__ rowptr, const int* __restrict__ csr_src,
                            const float* __restrict__ b, float* __restrict__ outh,
                            int Ctot, int H) {
  const int lanes = Ctot >> 2;
  int tid = blockIdx.x * blockDim.x + threadIdx.x;
  if (tid >= N_NODES * lanes) return;
  int dst = tid / lanes, lane = tid - dst * lanes;
  int c0 = lane << 2;
  int head = c0 >> 6;
  int s0 = rowptr[dst], s1 = rowptr[dst + 1];
  float ax = 0.f, ay = 0.f, az = 0.f, aw = 0.f;
  for (int p = s0; p < s1; ++p) {
    float a = alpha[p * H + head];
    const float4 hv = *(const float4*)(h + (size_t)csr_src[p] * Ctot + c0);
    ax += a * hv.x; ay += a * hv.y; az += a * hv.z; aw += a * hv.w;
  }
  const float4 bb = *(const float4*)(b + c0);
  ax += bb.x; ay += bb.y; az += bb.z; aw += bb.w;
  float4 o;
  o.x = (ax > 0.f) ? ax : (__expf(ax) - 1.f);
  o.y = (ay > 0.f) ? ay : (__expf(ay) - 1.f);
  o.z = (az > 0.f) ? az : (__expf(az) - 1.f);
  o.w = (aw > 0.f) ? aw : (__expf(aw) - 1.f);
  *(float4*)(outh + (size_t)dst * Ctot + c0) = o;
}

// ---------------- mean pool over graphs ----------------
__global__ void pool_sum(const float* __restrict__ h, const int* __restrict__ batch,
                         float* __restrict__ sums, float* __restrict__ cnt) {
  int tid = blockIdx.x * blockDim.x + threadIdx.x;
  if (tid >= N_NODES * HID) return;
  int node = tid >> 6, c = tid & 63;
  int g = batch[node];
  __hip_atomic_fetch_add(&sums[g * HID + c], h[tid],
                         __ATOMIC_RELAXED, __HIP_MEMORY_SCOPE_AGENT);
  if (c == 0)
    __hip_atomic_fetch_add(&cnt[g], 1.0f, __ATOMIC_RELAXED, __HIP_MEMORY_SCOPE_AGENT);
}
__global__ void pool_div(const float* __restrict__ sums, const float* __restrict__ cnt,
                         float* __restrict__ out) {
  int tid = blockIdx.x * blockDim.x + threadIdx.x;
  if (tid >= N_GRAPHS * HID) return;
  out[tid] = sums[tid] / fmaxf(cnt[tid >> 6], 1.0f);
}

// ---------------- host side ----------------
static inline int ceil_div(long long a, int b) { return (int)((a + b - 1) / b); }

extern "C" void kernel_launch(void* const* d_in, const int* in_sizes, int n_in,
                              void* d_out, int out_size, void* d_ws, size_t ws_size,
                              hipStream_t stream) {
  const float* x         = (const float*)d_in[0];
  const float* W1        = (const float*)d_in[1];
  const float* att_src1  = (const float*)d_in[2];
  const float* att_dst1  = (const float*)d_in[3];
  const float* b1        = (const float*)d_in[4];
  const float* W2        = (const float*)d_in[5];
  const float* att_src2  = (const float*)d_in[6];
  const float* att_dst2  = (const float*)d_in[7];
  const float* b2        = (const float*)d_in[8];
  const int*   ei        = (const int*)d_in[9];
  const int*   batch     = (const int*)d_in[10];
  float*       out       = (float*)d_out;

  // ---- workspace carve-out (256B aligned regions) ----
  char* wsb = (char*)d_ws;
  size_t off = 0;
  auto carve = [&](size_t bytes) -> void* {
    void* p = (void*)(wsb + off);
    off += (bytes + 255) & ~(size_t)255;
    return p;
  };
  float* h1      = (float*)carve((size_t)N_NODES * C1 * 4);   // x @ W1
  float* h1agg   = (float*)carve((size_t)N_NODES * C1 * 4);   // layer-1 output (post-ELU)
  float* as1     = (float*)carve((size_t)N_NODES * H1 * 4);
  float* ad1     = (float*)carve((size_t)N_NODES * H1 * 4);
  float* alpha1  = (float*)carve((size_t)E_TOT   * H1 * 4);
  float* h2      = (float*)carve((size_t)N_NODES * HID * 4);  // h1act @ W2
  float* h2agg   = (float*)carve((size_t)N_NODES * HID * 4);
  float* as2     = (float*)carve((size_t)N_NODES * 4);
  float* ad2     = (float*)carve((size_t)N_NODES * 4);
  float* alpha2  = (float*)carve((size_t)E_TOT * 4);
  float* sums    = (float*)carve((size_t)N_GRAPHS * HID * 4);
  float* cnt     = (float*)carve((size_t)N_GRAPHS * 4);
  int*   deg     = (int*)carve((size_t)N_NODES * 4);
  int*   rowptr  = (int*)carve((size_t)(N_NODES + 1) * 4);
  int*   cursor  = (int*)carve((size_t)N_NODES * 4);
  int*   csr_srcb= (int*)carve((size_t)E_TOT * 4);
  int*   bsum    = (int*)carve((size_t)1024 * 4);

  const int TPB = 256;
  const int NB_SCAN = ceil_div(N_NODES, SCAN_B);  // 196

  // ---- init ----
  fill_i32<<<ceil_div(N_NODES, TPB), TPB, 0, stream>>>(deg, N_NODES, 0);
  fill_f32<<<ceil_div(N_GRAPHS * HID, TPB), TPB, 0, stream>>>(sums, N_GRAPHS * HID, 0.f);
  fill_f32<<<1, 64, 0, stream>>>(cnt, N_GRAPHS, 0.f);

  // ---- build CSR by destination ----
  csr_deg<<<ceil_div(E_TOT, TPB), TPB, 0, stream>>>(ei, deg);
  scan1<<<NB_SCAN, SCAN_B, 0, stream>>>(deg, rowptr, bsum, N_NODES);
  scan2<<<1, 1024, 0, stream>>>(bsum, NB_SCAN);
  scan3<<<NB_SCAN, SCAN_B, 0, stream>>>(rowptr, bsum, N_NODES, E_TOT);
  copy_i32<<<ceil_div(N_NODES, TPB), TPB, 0, stream>>>(cursor, rowptr, N_NODES);
  csr_place<<<ceil_div(E_TOT, TPB), TPB, 0, stream>>>(ei, cursor, csr_srcb);

  // ================= layer 1 : GATConv(128 -> 64, heads=4) =================
  gemm_wmma_f32<<<N_NODES / 16, 16 * 32, 16 * IN_DIM * 4, stream>>>(x, W1, h1, IN_DIM, C1);
  node_attn<<<ceil_div(N_NODES * H1, TPB), TPB, 0, stream>>>(h1, att_src1, att_dst1, as1, ad1, H1);
  attn_softmax_csr<<<ceil_div(N_NODES * H1, TPB), TPB, 0, stream>>>(as1, ad1, rowptr, csr_srcb, alpha1, H1);
  agg_csr_elu<<<ceil_div((long long)N_NODES * (C1 / 4), TPB), TPB, 0, stream>>>(
      h1, alpha1, rowptr, csr_srcb, b1, h1agg, C1, H1);

  // ================= layer 2 : GATConv(256 -> 64, heads=1) =================
  gemm_wmma_f32<<<N_NODES / 16, 4 * 32, 16 * C1 * 4, stream>>>(h1agg, W2, h2, C1, HID);
  node_attn<<<ceil_div(N_NODES, TPB), TPB, 0, stream>>>(h2, att_src2, att_dst2, as2, ad2, 1);
  attn_softmax_csr<<<ceil_div(N_NODES, TPB), TPB, 0, stream>>>(as2, ad2, rowptr, csr_srcb, alpha2, 1);
  agg_csr_elu<<<ceil_div((long long)N_NODES * (HID / 4), TPB), TPB, 0, stream>>>(
      h2, alpha2, rowptr, csr_srcb, b2, h2agg, HID, 1);

  // ================= global mean pool =================
  pool_sum<<<ceil_div((long long)N_NODES * HID, TPB), TPB, 0, stream>>>(h2agg, batch, sums, cnt);
  pool_div<<<ceil_div(N_GRAPHS * HID, TPB), TPB, 0, stream>>>(sums, cnt, out);
}